// EGCN_4544075399689
// MI455X (gfx1250) — compile-verified
//
#include <hip/hip_runtime.h>
#include <hip/hip_bf16.h>
#include <cmath>

// ---------- problem constants (from reference) ----------
#define N_NODES   50000
#define N_EDGES   800000
#define IN_FEATS  256
#define H_FEATS   64
#define N_THETA   5     // OFFSET..D_POLY+OFFSET -> 5 polynomials
#define K_STEPS   6     // len(theta)-1 propagation steps
#define NEG_SLOPE 0.01f

typedef __attribute__((ext_vector_type(2))) float v2f;
typedef __attribute__((ext_vector_type(8))) float v8f;

struct ThetaCol { float v[N_THETA]; };   // one column theta[t][k] for fixed k

// ---------------- utility: zero a float buffer ----------------
__global__ void k_zero(float* __restrict__ p, int n) {
    int i = blockIdx.x * blockDim.x + threadIdx.x;
    if (i < n) p[i] = 0.0f;
}

// ---------------- degree: segment_sum(ones, src) ----------------
__global__ void k_degree(const int* __restrict__ src, float* __restrict__ deg) {
    int e = blockIdx.x * blockDim.x + threadIdx.x;
    if (e < N_EDGES) atomicAdd(&deg[src[e]], 1.0f);
}

// ---------------- d_inv = clip(deg,1)^-0.5 ----------------
__global__ void k_dinv(const float* __restrict__ deg, float* __restrict__ dinv) {
    int i = blockIdx.x * blockDim.x + threadIdx.x;
    if (i < N_NODES) {
        float d = deg[i];
        d = d > 1.0f ? d : 1.0f;
        dinv[i] = rsqrtf(d);
    }
}

// ---------------- h = leaky_relu(x @ W + b) via V_WMMA_F32_16X16X4_F32 ----
// One wave computes a 16(M) x 64(N) strip: A tile (16x4 f32) reused across the
// four 16x16 N-tiles. ISA layouts (CDNA5 ISA):
//   A 16x4 f32 : lanes 0-15 = rows 0-15 with K=0(VGPR0),K=1(VGPR1);
//                lanes 16-31 = rows with K=2,K=3.
//   B 4x16 f32 : lanes 0-15 = cols, K=0(VGPR0),K=1(VGPR1);
//                lanes 16-31 = cols, K=2,K=3.
//   C/D 16x16  : VGPR v -> M=v (lanes 0-15) / M=v+8 (lanes 16-31), N=lane&15.
__global__ void k_gemm_wmma(const float* __restrict__ x,
                            const float* __restrict__ W,
                            const float* __restrict__ bias,
                            float* __restrict__ h) {
    const int wave = (blockIdx.x * blockDim.x + threadIdx.x) >> 5;
    const int lane = threadIdx.x & 31;
    const int nWaves = N_NODES / 16;           // 3125, exact
    if (wave >= nWaves) return;                // wave-uniform: EXEC stays all-1s

    const int rowBase = wave * 16;
    const int half = lane >> 4;                // 0: K=0,1   1: K=2,3
    const int r    = lane & 15;                // row (for A) / col (for B)

    v8f acc0 = {}, acc1 = {}, acc2 = {}, acc3 = {};

    for (int k = 0; k < IN_FEATS; k += 4) {
        v2f a;
        const float* xp = x + (size_t)(rowBase + r) * IN_FEATS + k + 2 * half;
        a.x = xp[0];
        a.y = xp[1];

        const float* wp = W + (size_t)(k + 2 * half) * H_FEATS + r;
        v2f b0, b1, b2, b3;
        b0.x = wp[0];      b0.y = wp[H_FEATS];
        b1.x = wp[16];     b1.y = wp[H_FEATS + 16];
        b2.x = wp[32];     b2.y = wp[H_FEATS + 32];
        b3.x = wp[48];     b3.y = wp[H_FEATS + 48];

        acc0 = __builtin_amdgcn_wmma_f32_16x16x4_f32(false, a, false, b0, (short)0, acc0, false, false);
        acc1 = __builtin_amdgcn_wmma_f32_16x16x4_f32(false, a, false, b1, (short)0, acc1, false, false);
        acc2 = __builtin_amdgcn_wmma_f32_16x16x4_f32(false, a, false, b2, (short)0, acc2, false, false);
        acc3 = __builtin_amdgcn_wmma_f32_16x16x4_f32(false, a, false, b3, (short)0, acc3, false, false);
    }

    const int mBase = rowBase + half * 8;
    v8f accs[4] = {acc0, acc1, acc2, acc3};
    #pragma unroll
    for (int nt = 0; nt < 4; ++nt) {
        const int col = nt * 16 + r;
        const float bv = bias[col];
        #pragma unroll
        for (int v = 0; v < 8; ++v) {
            float val = accs[nt][v] + bv;
            val = val > 0.0f ? val : NEG_SLOPE * val;
            h[(size_t)(mBase + v) * H_FEATS + col] = val;
        }
    }
}

// ---------------- out[i, t*64+j] = theta[t][0] * h[i,j] ----------------
__global__ void k_init_out(const float* __restrict__ h, float* __restrict__ out,
                           ThetaCol th) {
    int tid = blockIdx.x * blockDim.x + threadIdx.x;
    if (tid >= N_NODES * H_FEATS) return;
    int i = tid >> 6;
    int j = tid & 63;
    float hv = h[tid];
    #pragma unroll
    for (int t = 0; t < N_THETA; ++t)
        out[(size_t)i * (N_THETA * H_FEATS) + t * H_FEATS + j] = th.v[t] * hv;
}

// ---------------- msg[dst] += f[src] * dinv[src]  (scatter-add) -----------
// One wave per edge; lane handles features 2*lane, 2*lane+1 (float2 gather,
// two global f32 atomics). Coalesced 256B gather per wave.
__global__ void k_scatter(const int* __restrict__ src, const int* __restrict__ dst,
                          const float* __restrict__ dinv, const float* __restrict__ f,
                          float* __restrict__ msg) {
    int tid  = blockIdx.x * blockDim.x + threadIdx.x;
    int e    = tid >> 5;
    int lane = tid & 31;
    if (e >= N_EDGES) return;
    int s = src[e];
    int d = dst[e];
    float w = dinv[s];
    int j = lane * 2;
    const float2 v = *(const float2*)(f + (size_t)s * H_FEATS + j);
    float* mp = msg + (size_t)d * H_FEATS + j;
    atomicAdd(mp,     v.x * w);
    atomicAdd(mp + 1, v.y * w);
}

// -------- f' = f - msg*dinv ; out[:,t] += theta[t][k] * f' ----------------
__global__ void k_update(const float* __restrict__ f, const float* __restrict__ msg,
                         const float* __restrict__ dinv, float* __restrict__ fnext,
                         float* __restrict__ out, ThetaCol th) {
    int tid = blockIdx.x * blockDim.x + threadIdx.x;
    if (tid >= N_NODES * H_FEATS) return;
    int i = tid >> 6;
    int j = tid & 63;
    float fn = f[tid] - msg[tid] * dinv[i];
    fnext[tid] = fn;
    float* op = out + (size_t)i * (N_THETA * H_FEATS) + j;
    #pragma unroll
    for (int t = 0; t < N_THETA; ++t)
        op[t * H_FEATS] += th.v[t] * fn;
}

// ---------------- host: theta table (matches _calculate_theta2) -----------
static double host_comb(int n, int k) {
    double r = 1.0;
    for (int t = 0; t < k; ++t) r = r * (double)(n - t) / (double)(t + 1);
    return r;
}
static void compute_thetas(double th[N_THETA][K_STEPS + 1]) {
    const double e = 1.4;
    const int d = 4, OFF = 2;
    for (int idx = 0; idx < N_THETA; ++idx) {
        int i = OFF + idx;                    // i in [2..6]
        int m = d - i + OFF;                  // 6 - i
        // beta(i+1, d+1-i+OFF) = G(i+1)G(d+1-i+OFF)/G(d+2+OFF)
        double B = tgamma((double)(i + 1)) * tgamma((double)(d + 1 - i + OFF)) /
                   tgamma((double)(d + 2 + OFF));
        for (int j = 0; j <= K_STEPS; ++j) th[idx][j] = 0.0;
        for (int j = 0; j <= m; ++j) {
            double c = pow(1.0 / e, (double)i) * host_comb(m, j) *
                       pow(-1.0 / e, (double)j) / (e * B);
            th[idx][i + j] += c;
        }
    }
}

extern "C" void kernel_launch(void* const* d_in, const int* in_sizes, int n_in,
                              void* d_out, int out_size, void* d_ws, size_t ws_size,
                              hipStream_t stream) {
    const float* x    = (const float*)d_in[0];
    const float* W    = (const float*)d_in[1];
    const float* bias = (const float*)d_in[2];
    const int*   src  = (const int*)d_in[3];
    const int*   dst  = (const int*)d_in[4];
    float* out = (float*)d_out;

    // workspace layout (floats): deg | dinv | fA(h) | fB | msg
    float* wsf  = (float*)d_ws;
    float* deg  = wsf;
    float* dinv = deg + N_NODES;
    float* fA   = dinv + N_NODES;                 // holds h, then ping-pongs
    float* fB   = fA + (size_t)N_NODES * H_FEATS;
    float* msg  = fB + (size_t)N_NODES * H_FEATS;

    double th[N_THETA][K_STEPS + 1];
    compute_thetas(th);

    const int NF = N_NODES * H_FEATS;             // 3,200,000

    // degree + dinv
    k_zero<<<(N_NODES + 255) / 256, 256, 0, stream>>>(deg, N_NODES);
    k_degree<<<(N_EDGES + 255) / 256, 256, 0, stream>>>(src, deg);
    k_dinv<<<(N_NODES + 255) / 256, 256, 0, stream>>>(deg, dinv);

    // h = leaky_relu(x@W + b) via f32 WMMA (3125 waves, 8 waves/block)
    const int nWaves = N_NODES / 16;
    k_gemm_wmma<<<(nWaves + 7) / 8, 256, 0, stream>>>(x, W, bias, fA);

    // out init with theta[:,0]
    ThetaCol tc;
    for (int t = 0; t < N_THETA; ++t) tc.v[t] = (float)th[t][0];
    k_init_out<<<(NF + 255) / 256, 256, 0, stream>>>(fA, out, tc);

    // 6 shared propagation steps (theta-independent recursion)
    float* fcur = fA;
    float* fnext = fB;
    for (int k = 1; k <= K_STEPS; ++k) {
        k_zero<<<(NF + 255) / 256, 256, 0, stream>>>(msg, NF);
        k_scatter<<<(N_EDGES * 32 + 255) / 256, 256, 0, stream>>>(src, dst, dinv, fcur, msg);
        for (int t = 0; t < N_THETA; ++t) tc.v[t] = (float)th[t][k];
        k_update<<<(NF + 255) / 256, 256, 0, stream>>>(fcur, msg, dinv, fnext, out, tc);
        float* tmp = fcur; fcur = fnext; fnext = tmp;
    }
}